// BiasEncoder_42064909697835
// MI455X (gfx1250) — compile-verified
//
#include <hip/hip_runtime.h>

typedef __attribute__((ext_vector_type(2))) float v2f;
typedef __attribute__((ext_vector_type(8))) float v8f;

#define MHOP 5
#define HD   8
#define NN   128
#define BATCH 32
#define NP1  129

// ---------------------------------------------------------------------------
// Kernel 1: padded border (row 0 / col 0) + graph token.
// out[b,j,0,0] = 0 ; out[b,j,i,0] = t[j] (i>=1) ; out[b,j,0,m] = t[j] (m>=1)
// ---------------------------------------------------------------------------
__global__ __launch_bounds__(256)
void bias_border_kernel(const float* __restrict__ graph_token,
                        float* __restrict__ out) {
  int tid = blockIdx.x * blockDim.x + threadIdx.x;
  const int total = BATCH * HD * NP1;
  if (tid >= total) return;
  int b = tid / (HD * NP1);
  int r = tid % (HD * NP1);
  int j = r / NP1;
  int i = r % NP1;
  float t = graph_token[j];
  float* base = out + (size_t)(b * HD + j) * (NP1 * NP1);
  base[i] = (i == 0) ? 0.0f : t;            // row 0
  if (i > 0) base[(size_t)i * NP1] = t;     // col 0
}

// ---------------------------------------------------------------------------
// Kernel 2: main. One wave32 handles a tile of 16 pairs p=(b,n,m).
// edge_bias = (E[16x40]) x (W[40x16]) via 10x V_WMMA_F32_16X16X4_F32,
// then fused hop-divide + spatial bias + scatter-store into padded output.
// A 16x4 f32 layout: lanes 0-15 -> K=k0,k0+1 ; lanes 16-31 -> K=k0+2,k0+3.
// B 4x16 f32 layout: VGPR0 = row K=k0 (lanes 0-15) / K=k0+2 (lanes 16-31),
//                    VGPR1 = row K=k0+1 / K=k0+3.
// C/D 16x16 f32:     VGPR v, lane L<16 -> (M=v, N=L); L>=16 -> (M=v+8, N=L-16)
// ---------------------------------------------------------------------------
__global__ __launch_bounds__(256)
void bias_main_kernel(const int* __restrict__ spatial_pos,
                      const int* __restrict__ edge_input,
                      const float* __restrict__ spatial_w,
                      const float* __restrict__ edge_w,
                      const float* __restrict__ edge_dis_w,
                      float* __restrict__ out) {
  const int lane    = threadIdx.x & 31;       // wave32
  const int wave    = threadIdx.x >> 5;       // 0..7 waves per block
  const int tile    = blockIdx.x * 8 + wave;  // 16 pairs per tile
  const int pb      = tile * 16;

  const int  n16     = lane & 15;             // N column (j) / A row (pair)
  const bool hi_half = lane >= 16;
  const int  ko      = hi_half ? 2 : 0;       // K offset within chunk

  // ---- B fragments: W = edge_dis_w[d][h][j] reshaped to 40x16 (cols 8..15 = 0)
  v2f bfrag[10];
#pragma unroll
  for (int c = 0; c < 10; ++c) {
    const int d  = c >> 1;
    const int hb = (c & 1) * 4;
    float bx = 0.0f, by = 0.0f;
    if (n16 < HD) {
      bx = edge_dis_w[d * 64 + (hb + ko)     * HD + n16];
      by = edge_dis_w[d * 64 + (hb + ko + 1) * HD + n16];
    }
    bfrag[c].x = bx;
    bfrag[c].y = by;
  }

  v8f acc = {};
  const int prow = pb + n16;                  // pair owning this lane's A rows
  const float third = 1.0f / 3.0f;

#pragma unroll
  for (int d = 0; d < MHOP; ++d) {
    // three F-indices for (pair, d); edge_input is (B,N,N,8,3)
    const int* ep = edge_input + ((size_t)prow * 8 + d) * 3;
    int i0 = ep[0]; i0 = i0 < 0 ? 0 : i0;
    int i1 = ep[1]; i1 = i1 < 0 ? 0 : i1;
    int i2 = ep[2]; i2 = i2 < 0 ? 0 : i2;
    // edge_w rows (H=8 floats); h-pairs {ko,ko+1} and {ko+4,ko+5}: 8B aligned
    v2f w0l = *(const v2f*)(edge_w + i0 * HD + ko);
    v2f w1l = *(const v2f*)(edge_w + i1 * HD + ko);
    v2f w2l = *(const v2f*)(edge_w + i2 * HD + ko);
    v2f w0h = *(const v2f*)(edge_w + i0 * HD + ko + 4);
    v2f w1h = *(const v2f*)(edge_w + i1 * HD + ko + 4);
    v2f w2h = *(const v2f*)(edge_w + i2 * HD + ko + 4);

    v2f alo, ahi;                              // E = mean over F of gathers
    alo.x = (w0l.x + w1l.x + w2l.x) * third;   // h = hb+ko
    alo.y = (w0l.y + w1l.y + w2l.y) * third;   // h = hb+ko+1
    ahi.x = (w0h.x + w1h.x + w2h.x) * third;   // h = hb+ko+4
    ahi.y = (w0h.y + w1h.y + w2h.y) * third;

    // chunk 2d   : K = d*8 + {0..3}  (h 0..3 of hop d)
    acc = __builtin_amdgcn_wmma_f32_16x16x4_f32(
        false, alo, false, bfrag[2 * d], (short)0, acc, false, false);
    // chunk 2d+1 : K = d*8 + {4..7}
    acc = __builtin_amdgcn_wmma_f32_16x16x4_f32(
        false, ahi, false, bfrag[2 * d + 1], (short)0, acc, false, false);
  }

  // ---- epilogue: divide by hop count, add spatial bias, scatter store ----
#pragma unroll
  for (int v = 0; v < 8; ++v) {
    const int p   = pb + v + (hi_half ? 8 : 0);   // pair for row M of acc[v]
    const int sp0 = spatial_pos[p];
    int s = (sp0 == 0) ? 1 : sp0;
    s = (s > 1) ? s - 1 : s;
    s = (s > MHOP) ? MHOP : s;
    const float inv = 1.0f / ((float)s + 1e-9f);
    if (n16 < HD) {
      const float val = spatial_w[sp0 * HD + n16] + acc[v] * inv;
      const int b  = p >> 14;           // p = ((b*128)+n)*128 + m
      const int nm = p & 16383;
      const int n  = nm >> 7;
      const int m  = nm & 127;
      out[((size_t)(b * HD + n16) * NP1 + (n + 1)) * NP1 + (m + 1)] = val;
    }
  }
}

// ---------------------------------------------------------------------------
extern "C" void kernel_launch(void* const* d_in, const int* in_sizes, int n_in,
                              void* d_out, int out_size, void* d_ws, size_t ws_size,
                              hipStream_t stream) {
  const int*   spatial_pos = (const int*)  d_in[0];
  const int*   edge_input  = (const int*)  d_in[1];
  const float* spatial_w   = (const float*)d_in[2];
  const float* edge_w      = (const float*)d_in[3];
  const float* edge_dis_w  = (const float*)d_in[4];
  const float* graph_token = (const float*)d_in[5];
  float* out = (float*)d_out;

  (void)in_sizes; (void)n_in; (void)out_size; (void)d_ws; (void)ws_size;

  // borders: 32*8*129 threads
  {
    const int total  = BATCH * HD * NP1;
    const int blocks = (total + 255) / 256;
    bias_border_kernel<<<blocks, 256, 0, stream>>>(graph_token, out);
  }
  // main: P = 32*128*128 pairs, 16 per wave, 8 waves per 256-thread block
  {
    const int P      = BATCH * NN * NN;     // 524288
    const int tiles  = P / 16;              // 32768 wave-tiles
    const int blocks = tiles / 8;           // 4096 blocks
    bias_main_kernel<<<blocks, 256, 0, stream>>>(
        spatial_pos, edge_input, spatial_w, edge_w, edge_dis_w, out);
  }
}